// PSCANet_28561532519144
// MI455X (gfx1250) — compile-verified
//
#include <hip/hip_runtime.h>
#include <math.h>

// ---------------- problem constants (match reference) ----------------
#define Bb_   64
#define Nn_   1024
#define Ll_   128
#define Mm_   64
#define NLAYER 20
#define SIGMA2_ 0.1f
#define EPS__   1e-5f

// ---------------- CDNA5 WMMA fp32 16x16x4 ----------------
typedef __attribute__((ext_vector_type(2))) float v2f;
typedef __attribute__((ext_vector_type(8))) float v8f;

__device__ __forceinline__ v8f wmma_k4(v2f a, v2f b, v8f c) {
  // D = A(16x4) * B(4x16) + C(16x16), fp32, wave32
  return __builtin_amdgcn_wmma_f32_16x16x4_f32(false, a, false, b, (short)0, c,
                                               false, false);
}

__device__ __forceinline__ v2f ld2(const float* p) {
  const float2 t = *reinterpret_cast<const float2*>(p);  // 8B aligned (even k)
  v2f r; r[0] = t.x; r[1] = t.y; return r;
}

// =====================================================================
// colnorm: norm[n] = sqrt(sum_l |S[l,n]|^2 / L);  Sn = S / norm
// =====================================================================
__global__ void colnorm_kernel(const float* __restrict__ Sr,
                               const float* __restrict__ Si,
                               float* __restrict__ Snr,
                               float* __restrict__ Sni) {
  const int n = blockIdx.x * blockDim.x + threadIdx.x;
  if (n >= Nn_) return;
  float s = 0.f;
  for (int l = 0; l < Ll_; ++l) {
    const float r = Sr[l * Nn_ + n], i = Si[l * Nn_ + n];
    s += r * r + i * i;
  }
  const float inv = 1.0f / sqrtf(s / (float)Ll_);
  for (int l = 0; l < Ll_; ++l) {
    Snr[l * Nn_ + n] = Sr[l * Nn_ + n] * inv;
    Sni[l * Nn_ + n] = Si[l * Nn_ + n] * inv;
  }
}

// ---------------- tiny elementwise kernels ----------------
__global__ void ew_ag_kernel(const float* __restrict__ a_,
                             const float* __restrict__ g,
                             float* __restrict__ ag) {
  const int i = blockIdx.x * blockDim.x + threadIdx.x;
  if (i < Bb_ * Nn_) ag[i] = a_[i] * sqrtf(g[i]);
}
__global__ void ew_w_kernel(const float* __restrict__ a,
                            const float* __restrict__ g,
                            float* __restrict__ w) {
  const int i = blockIdx.x * blockDim.x + threadIdx.x;
  if (i < Bb_ * Nn_) w[i] = a[i] * g[i];
}
__global__ void zero_kernel(float* __restrict__ p, int n) {
  const int i = blockIdx.x * blockDim.x + threadIdx.x;
  if (i < n) p[i] = 0.0f;
}

// =====================================================================
// Batched COMPLEX GEMM via V_WMMA_F32_16X16X4_F32, 2x2 register blocking:
// one wave computes a 32x32 complex output tile (16 v8f accumulators).
// All leading dims and K are template constants so the K-loop lowers to
// base-pointer + immediate-offset loads with zero per-iteration VALU.
//   out = alpha * (A') (B'')  [+ C] [+ diagAdd on real diagonal]
//   A'[m][k]  = A[m][k] * (HASW ? Wv[k] : 1)        (per-batch Wv)
//   B''[k][c] = TRANSB ? B[c][k] : B[k][c]          (conj handled by signs)
//   real = accRR + sR*accII ; imag = accIR + sI*accRI
//   mul (A*B):  sR=-1, sI=+1 ;  mulH (A*B^H): sR=+1, sI=-1
// grid = (N/32, M/32, batch), block = 32 (one wave).
// =====================================================================
template <bool TRANSB, bool HASW, bool HASC, int LDA, int LDB, int LDD, int K>
__global__ __launch_bounds__(32) void cgemm_wmma(
    const float* __restrict__ Ar, const float* __restrict__ Ai, long long sA,
    const float* __restrict__ Br, const float* __restrict__ Bi, long long sB,
    const float* __restrict__ Wv, long long sW,
    const float* __restrict__ Cr, const float* __restrict__ Ci, long long sC,
    float* __restrict__ Dr, float* __restrict__ Di, long long sD,
    float alpha, float sR, float sI, float diagAdd) {
  const int tid = threadIdx.x;          // 0..31
  const int idx = tid & 15;
  const int half = tid >> 4;            // lanes 0-15: K pair 0/1 ; 16-31: 2/3
  const int n0 = blockIdx.x * 32;
  const int m0 = blockIdx.y * 32;
  const int b = blockIdx.z;

  const float* ar = Ar + b * sA;
  const float* ai = Ai + b * sA;
  const float* br = Br + b * sB;
  const float* bi = Bi + b * sB;

  // per-lane base pointers: fold row/col and the 2*half K-pair offset in
  const float* pAr[2]; const float* pAi[2];
  const float* pBr[2]; const float* pBi[2];
#pragma unroll
  for (int t = 0; t < 2; ++t) {
    const int m = m0 + 16 * t + idx;
    const int c = n0 + 16 * t + idx;
    pAr[t] = ar + m * LDA + 2 * half;
    pAi[t] = ai + m * LDA + 2 * half;
    if (TRANSB) {
      pBr[t] = br + c * LDB + 2 * half;
      pBi[t] = bi + c * LDB + 2 * half;
    } else {
      pBr[t] = br + c + 2 * half * LDB;
      pBi[t] = bi + c + 2 * half * LDB;
    }
  }
  const float* pW = HASW ? (Wv + b * sW + 2 * half) : nullptr;

  v8f accRR[2][2] = {}, accII[2][2] = {}, accIR[2][2] = {}, accRI[2][2] = {};

#pragma unroll 4
  for (int k0 = 0; k0 < K; k0 += 4) {
    v2f fw;
    if (HASW) fw = ld2(pW + k0);
    v2f fAr[2], fAi[2], fBr[2], fBi[2];
#pragma unroll
    for (int t = 0; t < 2; ++t) {
      fAr[t] = ld2(pAr[t] + k0);
      fAi[t] = ld2(pAi[t] + k0);
      if (HASW) { fAr[t] *= fw; fAi[t] *= fw; }
      if (TRANSB) {
        fBr[t] = ld2(pBr[t] + k0);
        fBi[t] = ld2(pBi[t] + k0);
      } else {
        fBr[t][0] = pBr[t][k0 * LDB];
        fBr[t][1] = pBr[t][(k0 + 1) * LDB];
        fBi[t][0] = pBi[t][k0 * LDB];
        fBi[t][1] = pBi[t][(k0 + 1) * LDB];
      }
    }
#pragma unroll
    for (int mi = 0; mi < 2; ++mi) {
#pragma unroll
      for (int ni = 0; ni < 2; ++ni) {
        accRR[mi][ni] = wmma_k4(fAr[mi], fBr[ni], accRR[mi][ni]);
        accII[mi][ni] = wmma_k4(fAi[mi], fBi[ni], accII[mi][ni]);
        accIR[mi][ni] = wmma_k4(fAi[mi], fBr[ni], accIR[mi][ni]);
        accRI[mi][ni] = wmma_k4(fAr[mi], fBi[ni], accRI[mi][ni]);
      }
    }
  }

#pragma unroll
  for (int mi = 0; mi < 2; ++mi) {
#pragma unroll
    for (int ni = 0; ni < 2; ++ni) {
#pragma unroll
      for (int r = 0; r < 8; ++r) {
        const int row = m0 + 16 * mi + r + 8 * half;
        const int col = n0 + 16 * ni + idx;
        const long long o = (long long)row * LDD + col;
        float vr = alpha * (accRR[mi][ni][r] + sR * accII[mi][ni][r]);
        float vi = alpha * (accIR[mi][ni][r] + sI * accRI[mi][ni][r]);
        if (HASC) { vr += Cr[b * sC + o]; vi += Ci[b * sC + o]; }
        if (row == col) vr += diagAdd;
        Dr[b * sD + o] = vr;
        Di[b * sD + o] = vi;
      }
    }
  }
}

// =====================================================================
// Batched REAL GEMM (row-major, compile-time LD/K): D = alpha*A@B + beta*C
// 1x1 tile per wave (keeps 4096 waves in flight for the L^3 chain).
// =====================================================================
template <int LD, int K>
__global__ __launch_bounds__(32) void sgemm_wmma(
    const float* __restrict__ A, const float* __restrict__ B,
    const float* __restrict__ C, float* __restrict__ D,
    long long sMat, float alpha, float beta) {
  const int tid = threadIdx.x, idx = tid & 15, half = tid >> 4;
  const int n0 = blockIdx.x * 16, m0 = blockIdx.y * 16, b = blockIdx.z;
  const float* pA = A + b * sMat + (m0 + idx) * LD + 2 * half;
  const float* pB = B + b * sMat + (n0 + idx) + 2 * half * LD;
  v8f acc = {};
#pragma unroll 4
  for (int k0 = 0; k0 < K; k0 += 4) {
    v2f fa = ld2(pA + k0);
    v2f fb;
    fb[0] = pB[k0 * LD];
    fb[1] = pB[(k0 + 1) * LD];
    acc = wmma_k4(fa, fb, acc);
  }
#pragma unroll
  for (int r = 0; r < 8; ++r) {
    const int row = m0 + r + 8 * half;
    const long long o = b * sMat + (long long)row * LD + (n0 + idx);
    float v = alpha * acc[r];
    if (C) v += beta * C[o];
    D[o] = v;
  }
}

// =====================================================================
// 128x128 inverse, Gauss-Jordan without pivoting (inputs SPD-ish),
// in-place in exactly 64KB LDS. One block (256 thr) per batch.
// =====================================================================
__global__ __launch_bounds__(256) void invert128_kernel(
    const float* __restrict__ In, float* __restrict__ Out) {
  __shared__ float As[Ll_ * Ll_];       // 64 KB (WGP has 320 KB)
  const int t = threadIdx.x;            // 0..255
  const int b = blockIdx.x;
  const float* src = In + (long long)b * Ll_ * Ll_;
  float* dst = Out + (long long)b * Ll_ * Ll_;
  for (int i = t; i < Ll_ * Ll_; i += 256) As[i] = src[i];
  __syncthreads();

  const int i = t >> 1;                 // my row (2 threads per row)
  const int jb = (t & 1) * 64;          // my half of the row
  for (int k = 0; k < Ll_; ++k) {
    const float f = As[i * Ll_ + k];    // old pivot-column entry of my row
    const float p = As[k * Ll_ + k];
    const float invp = 1.0f / p;
    __syncthreads();
    if (i == k) {                       // scale pivot row; A[k][k] <- 1/p
      for (int j = jb; j < jb + 64; ++j) {
        const float v = As[k * Ll_ + j] * invp;
        As[k * Ll_ + j] = (j == k) ? invp : v;
      }
    }
    __syncthreads();
    if (i != k) {                       // eliminate; A[i][k] <- -f/p
      for (int j = jb; j < jb + 64; ++j) {
        const float akj = As[k * Ll_ + j];
        const float v = As[i * Ll_ + j];
        As[i * Ll_ + j] = ((j == k) ? 0.0f : v) - f * akj;
      }
    }
    __syncthreads();
  }
  for (int q = t; q < Ll_ * Ll_; q += 256) dst[q] = As[q];
}

// =====================================================================
// Fused per-layer tail: for a (batch, 16-column) slab
//   SiS   = Sinv @ Sn[:,slab]        (complex, WMMA, tile -> LDS, + d1)
//   ShSiS = Sh   @ SiS               (complex, WMMA, B operand from LDS)
//   d2    = sum_l SiS .* ShSiS ;  grad/clamp/update of a (in place)
// Avoids materializing the 2x32MB SiS / ShSiS intermediates.
// block = 256 (8 waves, one 16-row tile each); grid = (N/16, B)
// =====================================================================
__global__ __launch_bounds__(256) void fused_update_kernel(
    const float* __restrict__ Sinvr, const float* __restrict__ Sinvi,
    const float* __restrict__ Snr, const float* __restrict__ Sni,
    const float* __restrict__ Shr, const float* __restrict__ Shi,
    const float* __restrict__ g, const float* __restrict__ gamma, int layer,
    float* __restrict__ a) {
  __shared__ float sisr[Ll_ * 16];      // 8 KB
  __shared__ float sisi[Ll_ * 16];      // 8 KB
  __shared__ float d1acc[16];
  __shared__ float d2acc[16];

  const int tid = threadIdx.x;
  const int lane = tid & 31;
  const int wave = tid >> 5;            // 0..7 -> row tile
  const int idx = lane & 15;
  const int half = lane >> 4;
  const int b = blockIdx.y;
  const int n0 = blockIdx.x * 16;
  const int m0 = wave * 16;

  if (tid < 16) { d1acc[tid] = 0.f; d2acc[tid] = 0.f; }
  __syncthreads();

  // ---- phase 1: SiS tile + d1 ----
  {
    const float* pAr = Sinvr + (long long)b * Ll_ * Ll_ + (m0 + idx) * Ll_ + 2 * half;
    const float* pAi = Sinvi + (long long)b * Ll_ * Ll_ + (m0 + idx) * Ll_ + 2 * half;
    const float* pBr = Snr + (n0 + idx) + 2 * half * Nn_;
    const float* pBi = Sni + (n0 + idx) + 2 * half * Nn_;
    v8f aRR = {}, aII = {}, aIR = {}, aRI = {};
#pragma unroll 4
    for (int k0 = 0; k0 < Ll_; k0 += 4) {
      v2f far = ld2(pAr + k0);
      v2f fai = ld2(pAi + k0);
      v2f fbr, fbi;
      fbr[0] = pBr[k0 * Nn_];
      fbr[1] = pBr[(k0 + 1) * Nn_];
      fbi[0] = pBi[k0 * Nn_];
      fbi[1] = pBi[(k0 + 1) * Nn_];
      aRR = wmma_k4(far, fbr, aRR);
      aII = wmma_k4(fai, fbi, aII);
      aIR = wmma_k4(fai, fbr, aIR);
      aRI = wmma_k4(far, fbi, aRI);
    }
    float d1p = 0.f;
#pragma unroll
    for (int r = 0; r < 8; ++r) {
      const int row = m0 + r + 8 * half;          // l index
      const float sr = aRR[r] - aII[r];           // complex mul: Re
      const float si = aIR[r] + aRI[r];           // Im
      sisr[row * 16 + idx] = sr;
      sisi[row * 16 + idx] = si;
      d1p += Snr[row * Nn_ + n0 + idx] * sr + Sni[row * Nn_ + n0 + idx] * si;
    }
    atomicAdd(&d1acc[idx], d1p);
  }
  __syncthreads();

  // ---- phase 2: ShSiS tile + d2 (B operand streamed from LDS) ----
  {
    const float* pAr = Shr + (long long)b * Ll_ * Ll_ + (m0 + idx) * Ll_ + 2 * half;
    const float* pAi = Shi + (long long)b * Ll_ * Ll_ + (m0 + idx) * Ll_ + 2 * half;
    const float* pBr = sisr + idx + 2 * half * 16;
    const float* pBi = sisi + idx + 2 * half * 16;
    v8f aRR = {}, aII = {}, aIR = {}, aRI = {};
#pragma unroll 4
    for (int k0 = 0; k0 < Ll_; k0 += 4) {
      v2f far = ld2(pAr + k0);
      v2f fai = ld2(pAi + k0);
      v2f fbr, fbi;
      fbr[0] = pBr[k0 * 16];
      fbr[1] = pBr[(k0 + 1) * 16];
      fbi[0] = pBi[k0 * 16];
      fbi[1] = pBi[(k0 + 1) * 16];
      aRR = wmma_k4(far, fbr, aRR);
      aII = wmma_k4(fai, fbi, aII);
      aIR = wmma_k4(fai, fbr, aIR);
      aRI = wmma_k4(far, fbi, aRI);
    }
    float d2p = 0.f;
#pragma unroll
    for (int r = 0; r < 8; ++r) {
      const int row = m0 + r + 8 * half;
      const float hr = aRR[r] - aII[r];
      const float hi = aIR[r] + aRI[r];
      d2p += sisr[row * 16 + idx] * hr + sisi[row * 16 + idx] * hi;
    }
    atomicAdd(&d2acc[idx], d2p);
  }
  __syncthreads();

  if (tid < 16) {
    const long long o = (long long)b * Nn_ + n0 + tid;
    const float d1 = d1acc[tid];
    const float d2 = d2acc[tid];
    const float gv = g[o];
    const float av = a[o];
    const float grad = (d2 - d1 + EPS__) / (EPS__ + gv * d1 * d1);
    const float dd = fminf(1.0f - av, fmaxf(-av, grad));
    a[o] = av + gamma[layer] * dd;
  }
}

// =====================================================================
extern "C" void kernel_launch(void* const* d_in, const int* in_sizes, int n_in,
                              void* d_out, int out_size, void* d_ws,
                              size_t ws_size, hipStream_t stream) {
  (void)in_sizes; (void)n_in; (void)out_size; (void)ws_size;

  const float* a_in = (const float*)d_in[0];   // a_  [B,N]
  const float* g    = (const float*)d_in[1];   // g   [B,N]
  const float* H_r  = (const float*)d_in[2];   // [B,N,M]
  const float* H_i  = (const float*)d_in[3];
  const float* Z_r  = (const float*)d_in[4];   // [B,L,M]
  const float* Z_i  = (const float*)d_in[5];
  const float* S_r  = (const float*)d_in[6];   // [L,N]
  const float* S_i  = (const float*)d_in[7];
  const float* gamma = (const float*)d_in[8];  // [20]
  float* a = (float*)d_out;                    // [B,N] output (also state)

  // ---- workspace carve-up (~47 MB of floats) ----
  float* ws = (float*)d_ws;
  float* Snr = ws;  ws += Ll_ * Nn_;
  float* Sni = ws;  ws += Ll_ * Nn_;
  float* wbuf = ws; ws += Bb_ * Nn_;           // ag, then per-layer w
  float* Yr = ws;   ws += (long long)Bb_ * Ll_ * Mm_;
  float* Yi = ws;   ws += (long long)Bb_ * Ll_ * Mm_;
  float* Shr = ws;  ws += (long long)Bb_ * Ll_ * Ll_;
  float* Shi = ws;  ws += (long long)Bb_ * Ll_ * Ll_;
  float* Sgr = ws;  ws += (long long)Bb_ * Ll_ * Ll_;
  float* Sgi = ws;  ws += (long long)Bb_ * Ll_ * Ll_;
  float* Ainv = ws; ws += (long long)Bb_ * Ll_ * Ll_;
  float* U = ws;    ws += (long long)Bb_ * Ll_ * Ll_;
  float* Mx = ws;   ws += (long long)Bb_ * Ll_ * Ll_;
  float* Sir = ws;  ws += (long long)Bb_ * Ll_ * Ll_;
  float* V = ws;    ws += (long long)Bb_ * Ll_ * Ll_;
  float* Sii = ws;  ws += (long long)Bb_ * Ll_ * Ll_;

  const long long sLL = (long long)Ll_ * Ll_;
  const long long sLM = (long long)Ll_ * Mm_;
  const long long sNM = (long long)Nn_ * Mm_;
  const dim3 ew_grid((Bb_ * Nn_ + 255) / 256);

  // ---- setup ----
  colnorm_kernel<<<dim3((Nn_ + 255) / 256), 256, 0, stream>>>(S_r, S_i, Snr, Sni);
  ew_ag_kernel<<<ew_grid, 256, 0, stream>>>(a_in, g, wbuf);
  zero_kernel<<<ew_grid, 256, 0, stream>>>(a, Bb_ * Nn_);

  // Y = (Sn .* ag) @ H + Z       (complex mul: sR=-1, sI=+1)
  cgemm_wmma<false, true, true, Nn_, Mm_, Mm_, Nn_>
      <<<dim3(Mm_ / 32, Ll_ / 32, Bb_), 32, 0, stream>>>(
      Snr, Sni, 0,
      H_r, H_i, sNM,
      wbuf, Nn_,
      Z_r, Z_i, sLM,
      Yr, Yi, sLM,
      1.0f, -1.0f, 1.0f, 0.0f);

  // Sh = Y @ Y^H / M             (conj-transpose: sR=+1, sI=-1)
  cgemm_wmma<true, false, false, Mm_, Mm_, Ll_, Mm_>
      <<<dim3(Ll_ / 32, Ll_ / 32, Bb_), 32, 0, stream>>>(
      Yr, Yi, sLM,
      Yr, Yi, sLM,
      nullptr, 0,
      nullptr, nullptr, 0,
      Shr, Shi, sLL,
      1.0f / (float)Mm_, 1.0f, -1.0f, 0.0f);

  const dim3 gC(Ll_ / 32, Ll_ / 32, Bb_);
  const dim3 gS(Ll_ / 16, Ll_ / 16, Bb_);
  for (int layer = 0; layer < NLAYER; ++layer) {
    // w = a * g
    ew_w_kernel<<<ew_grid, 256, 0, stream>>>(a, g, wbuf);

    // Sigma = (Sn .* w) @ Sn^H + sigma2*I
    cgemm_wmma<true, true, false, Nn_, Nn_, Ll_, Nn_><<<gC, 32, 0, stream>>>(
        Snr, Sni, 0,
        Snr, Sni, 0,
        wbuf, Nn_,
        nullptr, nullptr, 0,
        Sgr, Sgi, sLL,
        1.0f, 1.0f, -1.0f, SIGMA2_);

    // complex inverse via real Schur decomposition
    invert128_kernel<<<Bb_, 256, 0, stream>>>(Sgr, Ainv);              // A_inv
    sgemm_wmma<Ll_, Ll_><<<gS, 32, 0, stream>>>(Sgi, Ainv, nullptr, U, sLL, 1.0f, 0.0f);
    sgemm_wmma<Ll_, Ll_><<<gS, 32, 0, stream>>>(U, Sgi, Sgr, Mx, sLL, 1.0f, 1.0f);
    invert128_kernel<<<Bb_, 256, 0, stream>>>(Mx, Sir);                // Sinv_r
    sgemm_wmma<Ll_, Ll_><<<gS, 32, 0, stream>>>(Ainv, Sgi, nullptr, V, sLL, 1.0f, 0.0f);
    sgemm_wmma<Ll_, Ll_><<<gS, 32, 0, stream>>>(V, Sir, nullptr, Sii, sLL, -1.0f, 0.0f); // Sinv_i

    // fused SiS / ShSiS / d1 / d2 / gradient update of a
    fused_update_kernel<<<dim3(Nn_ / 16, Bb_), 256, 0, stream>>>(
        Sir, Sii, Snr, Sni, Shr, Shi, g, gamma, layer, a);
  }
}